// Encoder_20289425506514
// MI455X (gfx1250) — compile-verified
//
#include <hip/hip_runtime.h>
#include <hip/hip_bf16.h>

typedef __attribute__((ext_vector_type(2))) float v2f;
typedef __attribute__((ext_vector_type(8))) float v8f;

#define IN_CH 128
#define HIDC  128
#define OUTC  64

// ---------------------------------------------------------------------------
// relaxed agent-scope fp32 atomic add -> global_atomic_add_f32 on gfx1250
// ---------------------------------------------------------------------------
__device__ inline void atomAddF(float* p, float v) {
    __hip_atomic_fetch_add(p, v, __ATOMIC_RELAXED, __HIP_MEMORY_SCOPE_AGENT);
}

// ---------------------------------------------------------------------------
// degree / normalization kernels
// ---------------------------------------------------------------------------
__global__ void deg_init_kernel(float* __restrict__ deg, int n) {
    int i = blockIdx.x * blockDim.x + threadIdx.x;
    if (i < n) deg[i] = 1.0f;                      // self-loop weight
}

__global__ void deg_accum_kernel(const int* __restrict__ dst,
                                 const float* __restrict__ w,
                                 float* __restrict__ deg, int n_edges) {
    int e = blockIdx.x * blockDim.x + threadIdx.x;
    if (e < n_edges) atomAddF(&deg[dst[e]], w[e]);
}

__global__ void deg_finalize_kernel(const float* __restrict__ deg,
                                    float* __restrict__ dinv, int n) {
    int i = blockIdx.x * blockDim.x + threadIdx.x;
    if (i < n) {
        float d = deg[i];
        dinv[i] = (d > 0.0f) ? rsqrtf(d) : 0.0f;
    }
}

// ---------------------------------------------------------------------------
// self-loop init: agg[i,f] = feat[i,f] * dinv[i]^2   (128-wide features)
// ---------------------------------------------------------------------------
__global__ void self_init_kernel(const float* __restrict__ feat,
                                 const float* __restrict__ dinv,
                                 float* __restrict__ out, long long total) {
    long long idx = (long long)blockIdx.x * blockDim.x + threadIdx.x;
    if (idx >= total) return;
    int i = (int)(idx >> 7);                       // idx / 128
    float di = dinv[i];
    out[idx] = feat[idx] * (di * di);
}

// ---------------------------------------------------------------------------
// edge scatter: one wave (32 lanes) per edge, 4 floats per lane (128-wide).
// Edge index forced into an SGPR so the 5 per-edge scalars become uniform
// loads instead of 32-lane redundant gathers.
// agg[dst] += feat[src] * (dinv[src] * w * dinv[dst])
// ---------------------------------------------------------------------------
__global__ void edge_scatter128_kernel(const float* __restrict__ feat,
                                       const int* __restrict__ src,
                                       const int* __restrict__ dst,
                                       const float* __restrict__ w,
                                       const float* __restrict__ dinv,
                                       float* __restrict__ out, int n_edges) {
    const int lane = threadIdx.x & 31;
    int e = __builtin_amdgcn_readfirstlane(
        blockIdx.x * (blockDim.x >> 5) + (threadIdx.x >> 5));
    if (e >= n_edges) return;                      // wave-uniform exit
    const int s  = src[e];
    const int d  = dst[e];
    const float nm = dinv[s] * w[e] * dinv[d];
    const float4 v = *(const float4*)(feat + (size_t)s * 128 + lane * 4);
    float* o = out + (size_t)d * 128 + lane * 4;
    atomAddF(o + 0, v.x * nm);
    atomAddF(o + 1, v.y * nm);
    atomAddF(o + 2, v.z * nm);
    atomAddF(o + 3, v.w * nm);
}

// ---------------------------------------------------------------------------
// WMMA f32 GEMM: C[M,N] = A[M,K] @ W[K,N] + bias (+relu),  N,K compile-time.
// block = 128 threads = 4 waves; each wave owns a 16x16 C tile.
// grid  = (ceil(M/16), N/64)
// A frag (16x4):  lane<16 -> K={k,k+1}; lane>=16 -> K={k+2,k+3}  (row = lane&15)
// B frag (4x16):  VGPR0 -> K = k + 2*(lane>>4); VGPR1 -> K+1     (col = lane&15)
// C frag (16x16): VGPR v -> row = v + 8*(lane>>4), col = lane&15
// ---------------------------------------------------------------------------
template <int N, int K, bool RELU>
__global__ void gemm_wmma_f32_kernel(const float* __restrict__ A,
                                     const float* __restrict__ W,
                                     const float* __restrict__ bias,
                                     float* __restrict__ C, int M) {
    const int lane    = threadIdx.x & 31;
    const int wave    = threadIdx.x >> 5;
    const int rowBase = blockIdx.x * 16;
    const int col     = blockIdx.y * 64 + wave * 16 + (lane & 15);
    const int half    = lane >> 4;                 // 0 or 1
    int row = rowBase + (lane & 15);
    row = (row < M) ? row : (M - 1);               // branchless clamp (loads only)

    // all inner-loop offsets are compile-time immediates
    const float* __restrict__ Arow = A + (size_t)row * K + half * 2;
    const float* __restrict__ Wp   = W + (size_t)(half * 2) * N + col;

    v8f acc = {};
    #pragma unroll
    for (int k = 0; k < K; k += 4) {
        v2f a = *(const v2f*)(Arow + k);           // 8B aligned
        v2f b;
        b.x = Wp[(size_t)k * N];
        b.y = Wp[(size_t)(k + 1) * N];
        acc = __builtin_amdgcn_wmma_f32_16x16x4_f32(
            /*neg_a=*/false, a, /*neg_b=*/false, b,
            /*c_mod=*/(short)0, acc, /*reuse_a=*/false, /*reuse_b=*/false);
    }

    const float bv = bias[col];
    #pragma unroll
    for (int v = 0; v < 8; ++v) {
        int r = rowBase + v + half * 8;
        if (r < M) {
            float val = acc[v] + bv;
            if (RELU) val = fmaxf(val, 0.0f);
            C[(size_t)r * N + col] = val;
        }
    }
}

// ---------------------------------------------------------------------------
// host launcher
// ---------------------------------------------------------------------------
extern "C" void kernel_launch(void* const* d_in, const int* in_sizes, int n_in,
                              void* d_out, int out_size, void* d_ws, size_t ws_size,
                              hipStream_t stream) {
    const float* x   = (const float*)d_in[0];   // [N, 128]
    const int*   ei  = (const int*)  d_in[1];   // [2, E]
    const float* w   = (const float*)d_in[2];   // [E]
    const float* W1  = (const float*)d_in[3];   // [128, 128]
    const float* b1  = (const float*)d_in[4];   // [128]
    const float* Wmu = (const float*)d_in[5];   // [128, 64]
    const float* bmu = (const float*)d_in[6];   // [64]
    const float* Wls = (const float*)d_in[7];   // [128, 64]
    const float* bls = (const float*)d_in[8];   // [64]

    const int n_nodes = in_sizes[0] / IN_CH;
    const int n_edges = in_sizes[2];
    const int* src = ei;
    const int* dst = ei + n_edges;

    // workspace layout
    float* deg  = (float*)d_ws;                         // [N]
    float* dinv = deg  + n_nodes;                       // [N]
    float* bufA = dinv + n_nodes;                       // [N,128] aggregation
    float* bufH = bufA + (size_t)n_nodes * HIDC;        // [N,128] hidden h

    float* out_mu = (float*)d_out;                      // [N,64]
    float* out_ls = out_mu + (size_t)n_nodes * OUTC;    // [N,64]

    const int T = 256;
    const long long nfeat = (long long)n_nodes * HIDC;
    const int blkNode  = (n_nodes + T - 1) / T;
    const int blkEdge  = (n_edges + T - 1) / T;
    const int blkFeat  = (int)((nfeat + T - 1) / T);
    const int blkScat  = (int)(((long long)n_edges * 32 + T - 1) / T);
    const int mTiles   = (n_nodes + 15) / 16;

    // ---- gcn_norm -----------------------------------------------------------
    deg_init_kernel<<<blkNode, T, 0, stream>>>(deg, n_nodes);
    deg_accum_kernel<<<blkEdge, T, 0, stream>>>(dst, w, deg, n_edges);
    deg_finalize_kernel<<<blkNode, T, 0, stream>>>(deg, dinv, n_nodes);

    // ---- layer 1: h = relu((A_n @ x) @ W1 + b1) -----------------------------
    self_init_kernel<<<blkFeat, T, 0, stream>>>(x, dinv, bufA, nfeat);
    edge_scatter128_kernel<<<blkScat, T, 0, stream>>>(x, src, dst, w, dinv, bufA, n_edges);
    gemm_wmma_f32_kernel<HIDC, IN_CH, true><<<dim3(mTiles, HIDC / 64), 128, 0, stream>>>(
        bufA, W1, b1, bufH, n_nodes);

    // ---- shared propagation: p = A_n @ h (one 128-wide pass for mu and ls) --
    self_init_kernel<<<blkFeat, T, 0, stream>>>(bufH, dinv, bufA, nfeat);
    edge_scatter128_kernel<<<blkScat, T, 0, stream>>>(bufH, src, dst, w, dinv, bufA, n_edges);

    // ---- heads: mu = p @ Wmu + bmu ; logstd = p @ Wls + bls -----------------
    gemm_wmma_f32_kernel<OUTC, HIDC, false><<<dim3(mTiles, OUTC / 64), 128, 0, stream>>>(
        bufA, Wmu, bmu, out_mu, n_nodes);
    gemm_wmma_f32_kernel<OUTC, HIDC, false><<<dim3(mTiles, OUTC / 64), 128, 0, stream>>>(
        bufA, Wls, bls, out_ls, n_nodes);
}